// ConnectedLossV5_83760452206650
// MI455X (gfx1250) — compile-verified
//
#include <hip/hip_runtime.h>

#define HW      (512 * 512)          // 2^18
#define NPIX    (16 * HW)            // 4,194,304 pixels
#define NBLK    (NPIX / 256)         // 16384 (merge/flatten grids)
#define NB_INIT (NPIX / 1024)        // 4096 blocks, 4 px/thread in k_init
#define NB_LOSS 512                  // grid-stride loss kernel
#define NBINS   4096
#define HISTSZ  (3 * NBINS)

typedef __attribute__((ext_vector_type(2))) float v2f;
typedef __attribute__((ext_vector_type(8))) float v8f;

// ---------------------------------------------------------------------------
// Wave-level exact-f32 sum of 32 lane values on the CDNA5 matrix pipe:
// D = A(16x4) * ones(4x16) + 0  ->  D[m][n] = rowsum(A,m).
// Requires EXEC == all ones (callers keep full 256-thread blocks, uniform flow).
// ---------------------------------------------------------------------------
__device__ __forceinline__ float wave_sum32(float v) {
  v2f a; a[0] = v;    a[1] = 0.0f;
  v2f b; b[0] = 1.0f; b[1] = 1.0f;   // ones B-matrix, layout-independent
  v8f c = {};
  c = __builtin_amdgcn_wmma_f32_16x16x4_f32(false, a, false, b,
                                            (short)0, c, false, false);
  float s = c[0] + c[1] + c[2] + c[3] + c[4] + c[5] + c[6] + c[7];
  return __shfl(s, 0, 32) + __shfl(s, 16, 32);
}

// Deterministic 256-thread block sum (8 waves). Valid on thread 0.
__device__ __forceinline__ float block_sum256(float v, float* lds8) {
  float w = wave_sum32(v);
  int lane = threadIdx.x & 31, wid = threadIdx.x >> 5;
  if (lane == 0) lds8[wid] = w;
  __syncthreads();
  float r = 0.0f;
  if (threadIdx.x == 0) {
    for (int i = 0; i < 8; ++i) r += lds8[i];
  }
  __syncthreads();            // lds8 reusable after return
  return r;
}

// torch-BCELoss-style clamped log
__device__ __forceinline__ float log_clamped(float x) {
  return (x > 0.0f) ? fmaxf(__logf(x), -100.0f) : -100.0f;
}

__device__ __forceinline__ float bce_term(float p, bool tm) {
  float tf = tm ? 1.0f : 0.0f;
  return -(tf * log_clamped(p) + (1.0f - tf) * log_clamped(1.0f - p));
}

// ---------------------------------------------------------------------------
// Lock-free union-find with min-hooking: parent pointers only decrease, so the
// final root of a component is its minimum flat index (reference label
// semantics). atomicMin makes the result order-independent (deterministic).
// ---------------------------------------------------------------------------
__device__ __forceinline__ int cc_find(const int* L, int x) {
  int p = L[x];
  while (p != x) { x = p; p = L[x]; }
  return x;
}

__device__ __forceinline__ void cc_union(int* L, int a, int b) {
  int u = cc_find(L, a);
  int v = cc_find(L, b);
  while (u != v) {
    if (v > u) { int t = u; u = v; v = t; }   // ensure u > v
    int old = atomicMin(&L[u], v);
    if (old == u) break;                      // hooked root u -> v
    u = old;                                  // merge remaining pair
  }
}

// small-state ints: [0..3] cnt[t]  [4..6] pref  [7..9] krem  [10..12] med

__global__ void k_zero(int* smalls) {
  int j = blockIdx.x * 256 + threadIdx.x;
  if (j < 64 + HISTSZ) smalls[j] = 0;
}

// argmax + background BCE partial + label init + target counts, 4 px/thread
__global__ void k_init(const float* __restrict__ pred, const int* __restrict__ tgt,
                       unsigned char* __restrict__ cls, int* __restrict__ labels,
                       float* __restrict__ phout, float* __restrict__ bgp,
                       int* __restrict__ smalls) {
  int g4 = (blockIdx.x * 256 + threadIdx.x) * 4;   // 16B-aligned pixel base
  int b  = g4 >> 18;
  int i  = g4 & (HW - 1);
  const float* p = pred + (((size_t)b * 4) << 18) + i;
  float4 q0 = *(const float4*)(p);
  float4 q1 = *(const float4*)(p + 1 * HW);
  float4 q2 = *(const float4*)(p + 2 * HW);
  float4 q3 = *(const float4*)(p + 3 * HW);
  int4   tv = *(const int4*)(tgt + g4);

  float pc0[4] = { q0.x, q0.y, q0.z, q0.w };
  float pc1[4] = { q1.x, q1.y, q1.z, q1.w };
  float pc2[4] = { q2.x, q2.y, q2.z, q2.w };
  float pc3[4] = { q3.x, q3.y, q3.z, q3.w };
  int   tva[4] = { tv.x & 3, tv.y & 3, tv.z & 3, tv.w & 3 };

  uchar4 cv;
  int4   lv;
  float4 pov;
  float  term = 0.0f;
  int    lcnt4[4] = { 0, 0, 0, 0 };
  unsigned char* cvp = (unsigned char*)&cv;
  int*   lvp  = (int*)&lv;
  float* povp = (float*)&pov;

#pragma unroll
  for (int e = 0; e < 4; ++e) {
    float best = pc0[e]; int c = 0;
    if (pc1[e] > best) { best = pc1[e]; c = 1; }
    if (pc2[e] > best) { best = pc2[e]; c = 2; }
    if (pc3[e] > best) { best = pc3[e]; c = 3; }
    cvp[e]  = (unsigned char)c;
    lvp[e]  = i + e;                 // self-root, plane-local index
    povp[e] = (c != 0) ? best : 0.0f;
    float pb = (c == 0) ? pc0[e] : 0.0f;
    term += bce_term(pb, tva[e] == 0);
    lcnt4[tva[e]] += 1;
  }

  *(uchar4*)(cls + g4)    = cv;
  *(int4*)(labels + g4)   = lv;
  *(float4*)(phout + g4)  = pov;

  __shared__ float lds8[8];
  __shared__ int   lcnt[4];
  float bs = block_sum256(term, lds8);
  if (threadIdx.x < 4) lcnt[threadIdx.x] = 0;
  __syncthreads();
#pragma unroll
  for (int t = 0; t < 4; ++t)
    if (lcnt4[t]) atomicAdd(&lcnt[t], lcnt4[t]);
  __syncthreads();
  if (threadIdx.x == 0) bgp[blockIdx.x] = bs;
  if (threadIdx.x < 4) atomicAdd(&smalls[threadIdx.x], lcnt[threadIdx.x]);
}

// one pass over right/down edges, class-gated 4-connectivity
__global__ void k_merge(const unsigned char* __restrict__ cls,
                        int* __restrict__ labels) {
  int g = blockIdx.x * 256 + threadIdx.x;
  int c = cls[g];
  if (c == 0) return;
  int b = g >> 18;
  int i = g & (HW - 1);
  int* L = labels + ((size_t)b << 18);
  int x = i & 511, y = i >> 9;
  if (x < 511 && cls[g + 1]   == c) cc_union(L, i, i + 1);
  if (y < 511 && cls[g + 512] == c) cc_union(L, i, i + 512);
}

// placeholder[g] = ((c-1)*B + b)*HW + min_index_in_component + 1
__global__ void k_flatten(const unsigned char* __restrict__ cls,
                          const int* __restrict__ labels,
                          int* __restrict__ ph) {
  int g = blockIdx.x * 256 + threadIdx.x;
  int c = cls[g];
  if (c == 0) { ph[g] = 0; return; }
  int b = g >> 18;
  int i = g & (HW - 1);
  int r = cc_find(labels + ((size_t)b << 18), i);
  ph[g] = ((((c - 1) * 16) + b) << 18) + r + 1;   // < 2^24
}

// 12-bit radix histogram, all 3 target labels in one sweep, LDS-privatized
__global__ void k_hist(const int* __restrict__ tgt, const int* __restrict__ ph,
                       int* __restrict__ hist, const int* __restrict__ smalls,
                       int pass) {
  __shared__ int lh[HISTSZ];
  for (int j = threadIdx.x; j < HISTSZ; j += 256) lh[j] = 0;
  __syncthreads();
  int base = blockIdx.x * 8192;                  // 512 blocks * 8192 = NPIX
  for (int k = 0; k < 32; ++k) {
    int g = base + k * 256 + threadIdx.x;
    __builtin_prefetch(&ph[g + 2048], 0, 1);     // global_prefetch_b8
    __builtin_prefetch(&tgt[g + 2048], 0, 1);
    int t = tgt[g];
    if (t >= 1) {
      int v = ph[g];
      if (pass == 0) {
        atomicAdd(&lh[((t - 1) << 12) + (v >> 12)], 1);
      } else if ((v >> 12) == smalls[3 + t]) {
        atomicAdd(&lh[((t - 1) << 12) + (v & 4095)], 1);
      }
    }
  }
  __syncthreads();
  for (int j = threadIdx.x; j < HISTSZ; j += 256) {
    int c = lh[j];
    if (c) atomicAdd(&hist[j], c);
  }
}

// scan 4096 bins per target; phase 0 -> prefix+remainder, phase 1 -> median.
// Also re-zeros the histogram for the next pass.
__global__ void k_select(int* __restrict__ smalls, int* __restrict__ hist,
                         int phase) {
  int tid = threadIdx.x;
  if (tid < 3) {
    int n = smalls[1 + tid];
    const int* h = hist + (tid << 12);
    if (n > 0) {
      if (phase == 0) {
        int k = (n - 1) >> 1;                    // torch lower-median rank
        int cum = 0, bin = 4095, rem = 0;
        for (int j = 0; j < NBINS; ++j) {
          int nc = cum + h[j];
          if (k < nc) { bin = j; rem = k - cum; break; }
          cum = nc;
        }
        smalls[4 + tid] = bin;
        smalls[7 + tid] = rem;
      } else {
        int k = smalls[7 + tid];
        int cum = 0, bin = 4095;
        for (int j = 0; j < NBINS; ++j) {
          int nc = cum + h[j];
          if (k < nc) { bin = j; break; }
          cum = nc;
        }
        smalls[10 + tid] = (smalls[4 + tid] << 12) | bin;
      }
    } else {
      smalls[4 + tid]  = -1;                     // prefix that never matches
      smalls[7 + tid]  = 0;
      smalls[10 + tid] = -1;                     // median that never matches
    }
  }
  __syncthreads();
  for (int j = tid; j < HISTSZ; j += 256) hist[j] = 0;
}

// fused c1/c2 partial sums for t = 1..3, grid-stride with register accumulators
__global__ void k_loss(const int* __restrict__ tgt, const int* __restrict__ ph,
                       const float* __restrict__ phout,
                       const int* __restrict__ smalls,
                       float* __restrict__ c1p, float* __restrict__ c2p) {
  int med[3] = { smalls[10], smalls[11], smalls[12] };
  float acc1[3] = { 0.0f, 0.0f, 0.0f };
  float acc2[3] = { 0.0f, 0.0f, 0.0f };
  int base = blockIdx.x * 8192;                  // 512 blocks * 8192 = NPIX
  for (int k = 0; k < 32; ++k) {
    int g = base + k * 256 + threadIdx.x;
    __builtin_prefetch(&ph[g + 2048], 0, 1);
    __builtin_prefetch(&phout[g + 2048], 0, 1);
    int   t  = tgt[g];
    int   v  = ph[g];
    float po = phout[g];
#pragma unroll
    for (int q = 0; q < 3; ++q) {
      bool tm = (t == q + 1);
      bool fm = (v == med[q]);
      acc1[q] += bce_term(fm ? po : 0.0f, tm);
      if (tm && !fm) acc2[q] += po;
    }
  }
  __shared__ float lds8[8];
#pragma unroll
  for (int q = 0; q < 3; ++q) {
    float s1 = block_sum256(acc1[q], lds8);
    float s2 = block_sum256(acc2[q], lds8);
    if (threadIdx.x == 0) {
      c1p[q * NB_LOSS + blockIdx.x] = s1;
      c2p[q * NB_LOSS + blockIdx.x] = s2;
    }
    __syncthreads();
  }
}

// single-block deterministic final combine
__global__ void k_final(const float* __restrict__ bgp,
                        const float* __restrict__ c1p,
                        const float* __restrict__ c2p,
                        const int* __restrict__ smalls,
                        float* __restrict__ out) {
  __shared__ float red[256];
  const float* arrs[7] = { bgp,
                           c1p, c1p + NB_LOSS, c1p + 2 * NB_LOSS,
                           c2p, c2p + NB_LOSS, c2p + 2 * NB_LOSS };
  const int lens[7] = { NB_INIT, NB_LOSS, NB_LOSS, NB_LOSS,
                        NB_LOSS, NB_LOSS, NB_LOSS };
  float sums[7];
  for (int a = 0; a < 7; ++a) {
    float s = 0.0f;
    for (int j = threadIdx.x; j < lens[a]; j += 256) s += arrs[a][j];
    red[threadIdx.x] = s;
    __syncthreads();
    for (int off = 128; off > 0; off >>= 1) {
      if (threadIdx.x < off) red[threadIdx.x] += red[threadIdx.x + off];
      __syncthreads();
    }
    sums[a] = red[0];
    __syncthreads();
  }
  if (threadIdx.x == 0) {
    float invN = 1.0f / (float)NPIX;
    float loss = sums[0] * invN;                       // background BCE mean
    for (int q = 0; q < 3; ++q) {
      int n = smalls[1 + q];
      if (n > 0) loss += sums[1 + q] * invN + sums[4 + q] / (float)n;
    }
    int nu = 0;
    for (int t = 0; t < 4; ++t) nu += (smalls[t] > 0) ? 1 : 0;
    out[0] = loss / ((float)nu * 2.0f + 1.0f);
  }
}

extern "C" void kernel_launch(void* const* d_in, const int* in_sizes, int n_in,
                              void* d_out, int out_size, void* d_ws, size_t ws_size,
                              hipStream_t stream) {
  (void)in_sizes; (void)n_in; (void)out_size; (void)ws_size;
  const float* pred = (const float*)d_in[0];    // [16,4,512,512] f32
  const int*   tgt  = (const int*)d_in[1];      // [16,1,512,512] i32
  float*       out  = (float*)d_out;            // scalar f32

  // workspace layout (~55 MB)
  int*           smalls = (int*)d_ws;                          // 64 ints
  int*           hist   = smalls + 64;                         // 3*4096 ints
  float*         bgp    = (float*)(hist + HISTSZ);             // NB_INIT
  float*         c1p    = bgp + NB_INIT;                       // 3*NB_LOSS
  float*         c2p    = c1p + 3 * NB_LOSS;                   // 3*NB_LOSS
  unsigned char* cls    = (unsigned char*)(c2p + 3 * NB_LOSS); // NPIX bytes
  int*           labels = (int*)(cls + NPIX);                  // NPIX ints
  int*           ph     = labels + NPIX;                       // NPIX ints
  float*         phout  = (float*)(ph + NPIX);                 // NPIX floats

  k_zero   <<<49,      256, 0, stream>>>(smalls);
  k_init   <<<NB_INIT, 256, 0, stream>>>(pred, tgt, cls, labels, phout, bgp, smalls);
  k_merge  <<<NBLK,    256, 0, stream>>>(cls, labels);
  k_flatten<<<NBLK,    256, 0, stream>>>(cls, labels, ph);
  k_hist   <<<512,     256, 0, stream>>>(tgt, ph, hist, smalls, 0);
  k_select <<<1,       256, 0, stream>>>(smalls, hist, 0);
  k_hist   <<<512,     256, 0, stream>>>(tgt, ph, hist, smalls, 1);
  k_select <<<1,       256, 0, stream>>>(smalls, hist, 1);
  k_loss   <<<NB_LOSS, 256, 0, stream>>>(tgt, ph, phout, smalls, c1p, c2p);
  k_final  <<<1,       256, 0, stream>>>(bgp, c1p, c2p, smalls, out);
}